// DiagonalPositiveLinear_50629074486108
// MI455X (gfx1250) — compile-verified
//
#include <hip/hip_runtime.h>
#include <math.h>

// out[b, f] = input[b, f] * exp(log_weight[f])
// input: 8192 x 4096 fp32 (row-major), log_weight: 4096 fp32, out: 8192 x 4096 fp32.
//
// Memory-bound streaming kernel: 256 MiB total traffic -> ~11.5 us floor at
// 23.3 TB/s. Strategy: b128 vector loads/stores with non-temporal hints,
// exp() amortized across 8 rows per thread, wave32-coalesced access pattern.

typedef float v4f __attribute__((ext_vector_type(4)));

#define NROWS 8192
#define NCOLS 4096
#define NV (NCOLS / 4)          // 1024 float4-groups per row
#define ROWS_PER_BLOCK 8
#define BLOCK_THREADS 256

__global__ __launch_bounds__(BLOCK_THREADS)
void diag_pos_linear_kernel(const v4f* __restrict__ in,
                            const float* __restrict__ logw,
                            v4f* __restrict__ out)
{
    // Column vec4 index: 0..NV-1. gridDim.x = NV / BLOCK_THREADS = 4.
    const int c4 = blockIdx.x * BLOCK_THREADS + threadIdx.x;

    // Weight vector is tiny (16 KB) and reused by every row-block: regular
    // (RT) cached load so it stays resident in WGP$/L2.
    const v4f lw = ((const v4f*)logw)[c4];

    v4f s;
    s.x = expf(lw.x);
    s.y = expf(lw.y);
    s.z = expf(lw.z);
    s.w = expf(lw.w);

    // This block owns rows [blockIdx.y*ROWS_PER_BLOCK, +ROWS_PER_BLOCK).
    long idx = (long)blockIdx.y * ROWS_PER_BLOCK * NV + c4;

#pragma unroll
    for (int r = 0; r < ROWS_PER_BLOCK; ++r) {
        // Streaming data: read-once / write-once -> non-temporal hints so the
        // 256 MiB stream does not evict useful L2 contents.
        v4f v = __builtin_nontemporal_load(&in[idx]);
        v *= s;
        __builtin_nontemporal_store(v, &out[idx]);
        idx += NV;
    }
}

extern "C" void kernel_launch(void* const* d_in, const int* in_sizes, int n_in,
                              void* d_out, int out_size, void* d_ws, size_t ws_size,
                              hipStream_t stream) {
    (void)in_sizes; (void)n_in; (void)out_size; (void)d_ws; (void)ws_size;

    const v4f*   in   = (const v4f*)d_in[0];     // fp32 input, 8192*4096
    const float* logw = (const float*)d_in[1];   // fp32 log_weight, 4096
    v4f*         out  = (v4f*)d_out;             // fp32 output

    dim3 grid(NV / BLOCK_THREADS, NROWS / ROWS_PER_BLOCK);  // (4, 1024)
    diag_pos_linear_kernel<<<grid, BLOCK_THREADS, 0, stream>>>(in, logw, out);
}